// lossx2_73967926772308
// MI455X (gfx1250) — compile-verified
//
#include <hip/hip_runtime.h>
#include <math.h>

// Problem constants (match reference)
#define Bfull   32
#define Cc      512
#define Hh      64
#define Ww      64
#define Kk      21
#define Rr      6
#define BGROUPS 8
#define BPG     (Bfull / BGROUPS)   // 4 batches per group

typedef __attribute__((ext_vector_type(2))) float v2f;
typedef __attribute__((ext_vector_type(8))) float v8f;

// ---------------------------------------------------------------------------
// Kernel A: direct ROI box sums. One wave per (b, c) plane, 8 waves / block.
// part[bg][k][c] = sum over the 4 batches in group bg of boxsum(b,k,c)/(area*B)
// Deterministic: every output element written by exactly one wave, batches
// accumulated in a fixed order. No atomics.
// ---------------------------------------------------------------------------
__global__ __launch_bounds__(256)
void roi_kernel(const float* __restrict__ f,     // [B][C][H][W]
                const float* __restrict__ pre,   // [B][K][2]
                float* __restrict__ part)        // [BGROUPS][K][C]
{
    const int ct   = blockIdx.x;              // channel tile (8 channels)
    const int bg   = blockIdx.y;              // batch group
    const int wid  = threadIdx.x >> 5;
    const int lane = threadIdx.x & 31;
    const int c    = ct * 8 + wid;
    const int wOff   = lane & 15;             // column within ROI window (<=12 used)
    const int rowPar = lane >> 4;             // 0/1: even/odd row of the box

    __shared__ int   s_hl[Kk], s_hr[Kk], s_wd[Kk], s_dw[Kk];
    __shared__ float s_inv[Kk];

    float outAcc[Kk];
#pragma unroll
    for (int k = 0; k < Kk; ++k) outAcc[k] = 0.0f;

    for (int bb = 0; bb < BPG; ++bb) {
        const int b = bg * BPG + bb;
        __syncthreads();                      // previous iteration's readers done
        if (threadIdx.x < Kk) {
            const int k = threadIdx.x;
            const float x = pre[(b * Kk + k) * 2 + 0];
            const float y = pre[(b * Kk + k) * 2 + 1];
            // clip then truncate, exactly like jnp.clip(...).astype(int32)
            const int hl = (int)fmaxf(x - (float)Rr, 0.0f);
            const int hr = (int)fminf(x + (float)Rr, 63.0f);
            const int wd = (int)fmaxf(y - (float)Rr, 0.0f);
            const int wu = (int)fminf(y + (float)Rr, 63.0f);
            s_hl[k] = hl; s_hr[k] = hr; s_wd[k] = wd; s_dw[k] = wu - wd;
            // reference divides by (hr-hl+1)*(wu-wd+1), then means over B
            s_inv[k] = 1.0f / ((float)((hr - hl + 1) * (wu - wd + 1)) * (float)Bfull);
        }
        __syncthreads();

        const float* plane = f + ((size_t)b * Cc + c) * (size_t)(Hh * Ww);
#pragma unroll
        for (int k = 0; k < Kk; ++k) {
            const int hl = s_hl[k], hr = s_hr[k];
            const int wd = s_wd[k], dw = s_dw[k];
            float acc = 0.0f;
            if (wOff < dw) {
                for (int h = hl + rowPar; h < hr; h += 2)
                    acc += plane[h * Ww + wd + wOff];
            }
            // butterfly reduction across the wave (wave32)
#pragma unroll
            for (int m = 16; m >= 1; m >>= 1) acc += __shfl_xor(acc, m, 32);
            outAcc[k] += acc * s_inv[k];
        }
    }

    if (lane == 0) {
#pragma unroll
        for (int k = 0; k < Kk; ++k)
            part[((size_t)bg * Kk + k) * Cc + c] = outAcc[k];
    }
}

// ---------------------------------------------------------------------------
// Kernel B0: fixed-order reduce of the batch-group partials + EMA with fea,
// producing the zero-padded 48x512 "total" matrix for the MMD GEMM.
// ---------------------------------------------------------------------------
__global__ __launch_bounds__(256)
void build_T(const float* __restrict__ part,   // [2][BGROUPS][K][C]
             const float* __restrict__ fea1,   // [K][C]
             const float* __restrict__ fea2,   // [K][C]
             float* __restrict__ Tpad)         // [48][512]
{
    const int idx = blockIdx.x * 256 + threadIdx.x;
    if (idx >= 48 * Cc) return;
    const int row = idx >> 9;
    const int c   = idx & (Cc - 1);
    float v = 0.0f;
    if (row < 2 * Kk) {
        const int feat = (row >= Kk) ? 1 : 0;
        const int k    = feat ? (row - Kk) : row;
        const float* p = part + (size_t)feat * BGROUPS * Kk * Cc;
        float s = 0.0f;
        for (int g = 0; g < BGROUPS; ++g)           // fixed order: deterministic
            s += p[((size_t)g * Kk + k) * Cc + c];
        const float* fea = feat ? fea2 : fea1;
        v = 0.999f * s + (1.0f - 0.999f) * fea[k * Cc + c];
    }
    Tpad[idx] = v;
}

// ---------------------------------------------------------------------------
// Kernel B: Gram matrix via V_WMMA_F32_16X16X4_F32 (9 waves, one 16x16 tile
// each), then the full RBF-MMD finish with deterministic reductions.
//
// Fragment layouts per CDNA5 ISA 7.12.2:
//   A 16x4 f32 : lane L holds M = L%16, K = {2*(L/16), 2*(L/16)+1}  -> float2
//   B 4x16 f32 : lane L holds N = L%16, K = {2*(L/16), 2*(L/16)+1}  -> float2
// Since B = T^T for a Gram matrix, both fragments are the same contiguous
// float2 load: Tpad[tile*16 + L%16][k0 + 2*(L/16)].
// ---------------------------------------------------------------------------
__global__ __launch_bounds__(288)
void mmd_kernel(const float* __restrict__ Tpad,  // [48][512]
                float* __restrict__ out)         // scalar
{
    __shared__ float G[48 * 48];
    __shared__ float red[16];
    __shared__ float sh_sum;

    const int tid  = threadIdx.x;
    const int wid  = tid >> 5;        // 0..8
    const int lane = tid & 31;
    const int half = lane >> 4;
    const int r    = lane & 15;

    {   // --- Gram tile (EXEC all ones: all 288 threads active) ---
        const int ti = wid / 3, tj = wid % 3;
        const float* Arow = Tpad + (size_t)(ti * 16 + r) * Cc;
        const float* Brow = Tpad + (size_t)(tj * 16 + r) * Cc;
        v8f acc = {0.f, 0.f, 0.f, 0.f, 0.f, 0.f, 0.f, 0.f};
        for (int k0 = 0; k0 < Cc; k0 += 4) {
            v2f a = *(const v2f*)(Arow + k0 + 2 * half);
            v2f b = *(const v2f*)(Brow + k0 + 2 * half);
            // (neg_a, A, neg_b, B, c_mod, C, reuse_a, reuse_b)
            acc = __builtin_amdgcn_wmma_f32_16x16x4_f32(
                    false, a, false, b, (short)0, acc, false, false);
        }
        // C/D layout: VGPR v holds M = v + 8*(lane/16), N = lane%16
#pragma unroll
        for (int v = 0; v < 8; ++v) {
            const int M = v + 8 * half;
            G[(ti * 16 + M) * 48 + (tj * 16 + r)] = acc[v];
        }
    }
    __syncthreads();

    // --- bandwidth = sum(L2) / (N^2 - N) / kernel_mul^(kernel_num//2) ---
    float ls = 0.0f;
    for (int p = tid; p < 42 * 42; p += 288) {
        const int i = p / 42, j = p - i * 42;
        ls += G[i * 48 + i] + G[j * 48 + j] - 2.0f * G[i * 48 + j];
    }
#pragma unroll
    for (int m = 16; m >= 1; m >>= 1) ls += __shfl_xor(ls, m, 32);
    if (lane == 0) red[wid] = ls;
    __syncthreads();
    if (tid == 0) {
        float s = 0.0f;
        for (int w = 0; w < 9; ++w) s += red[w];   // fixed order
        sh_sum = s;
    }
    __syncthreads();
    const float bw0 = sh_sum / (float)(42 * 42 - 42) * 0.25f;  // /2^(5//2)

    // --- signed kernel sum: mean(XX + YY - XY - YX) over 21x21 ---
    float lk = 0.0f;
    for (int p = tid; p < 42 * 42; p += 288) {
        const int i = p / 42, j = p - i * 42;
        const float l2 = G[i * 48 + i] + G[j * 48 + j] - 2.0f * G[i * 48 + j];
        float ks = 0.0f;
#pragma unroll
        for (int q = 0; q < 5; ++q)
            ks += expf(-l2 / (bw0 * (float)(1 << q)));
        const float w = ((i < Kk) == (j < Kk)) ? 1.0f : -1.0f;
        lk += w * ks;
    }
    __syncthreads();
#pragma unroll
    for (int m = 16; m >= 1; m >>= 1) lk += __shfl_xor(lk, m, 32);
    if (lane == 0) red[wid] = lk;
    __syncthreads();
    if (tid == 0) {
        float s = 0.0f;
        for (int w = 0; w < 9; ++w) s += red[w];   // fixed order
        out[0] = s / (float)(Kk * Kk);
    }
}

// ---------------------------------------------------------------------------
// Launch. ws layout: part[2][BGROUPS][K][C] (688 KB) then Tpad[48][512]
// (96 KB) -> ~0.79 MB total. All buffers fully written before read, so no
// zero-init pass is needed and replays are deterministic.
// ---------------------------------------------------------------------------
extern "C" void kernel_launch(void* const* d_in, const int* in_sizes, int n_in,
                              void* d_out, int out_size, void* d_ws, size_t ws_size,
                              hipStream_t stream) {
    const float* f1   = (const float*)d_in[0];
    const float* f2   = (const float*)d_in[1];
    const float* pre1 = (const float*)d_in[2];
    const float* pre2 = (const float*)d_in[3];
    const float* fea1 = (const float*)d_in[4];
    const float* fea2 = (const float*)d_in[5];

    float* part = (float*)d_ws;                              // [2][BGROUPS][K][C]
    float* Tpad = part + (size_t)2 * BGROUPS * Kk * Cc;      // [48][512]

    dim3 gA(Cc / 8, BGROUPS, 1);
    roi_kernel<<<gA, 256, 0, stream>>>(f1, pre1, part);
    roi_kernel<<<gA, 256, 0, stream>>>(f2, pre2, part + (size_t)BGROUPS * Kk * Cc);

    build_T<<<(48 * Cc + 255) / 256, 256, 0, stream>>>(part, fea1, fea2, Tpad);

    mmd_kernel<<<1, 288, 0, stream>>>(Tpad, (float*)d_out);
}